// AdMSoftmaxLoss_69552700391485
// MI455X (gfx1250) — compile-verified
//
#include <hip/hip_runtime.h>
#include <hip/hip_bf16.h>

// AM-Softmax loss, MI455X (gfx1250), wave32.
// Kernel 1: one workgroup per row, NT float4 streaming loads + v_exp_f32,
//           LDS tree reduction, per-row loss -> d_ws.
// Kernel 2: one wave, 64 chained V_WMMA_F32_16X16X4_F32 (B = ones) to sum
//           4096 row losses in fp32, then write -mean.

#define NROWS 4096
#define NCOLS 20000
#define SCALE 30.0f
#define MARGIN 0.4f
#define LOG2E 1.44269504088896340736f
#define LN2   0.69314718055994530942f

typedef float v2f __attribute__((ext_vector_type(2)));
typedef float v4f __attribute__((ext_vector_type(4)));
typedef float v8f __attribute__((ext_vector_type(8)));

__global__ __launch_bounds__(256) void adms_row_kernel(
    const float* __restrict__ wf, const int* __restrict__ labels,
    float* __restrict__ rowL) {
  const int row = blockIdx.x;
  const float* base = wf + (size_t)row * NCOLS;
  const v4f* vec = (const v4f*)base;            // 20000 % 4 == 0 -> 5000 vec4
  const float k = SCALE * LOG2E;                // exp(S*x) = exp2(k*x)

  float acc = 0.0f;
  for (int i = threadIdx.x; i < NCOLS / 4; i += 256) {
    // gfx1250 prefetch one iteration ahead (global_prefetch_b8)
    __builtin_prefetch((const char*)(vec + i) + 256 * sizeof(v4f), 0, 0);
    // stream past L2: 327MB read-once >> 192MB L2 -> non-temporal
    v4f v = __builtin_nontemporal_load(vec + i);
    acc += __builtin_amdgcn_exp2f(v.x * k);
    acc += __builtin_amdgcn_exp2f(v.y * k);
    acc += __builtin_amdgcn_exp2f(v.z * k);
    acc += __builtin_amdgcn_exp2f(v.w * k);
  }

  __shared__ float red[256];
  red[threadIdx.x] = acc;
  __syncthreads();
  for (int off = 128; off > 0; off >>= 1) {
    if (threadIdx.x < (unsigned)off) red[threadIdx.x] += red[threadIdx.x + off];
    __syncthreads();
  }

  if (threadIdx.x == 0) {
    const int lbl = labels[row];
    const float t = base[lbl];
    const float et = __builtin_amdgcn_exp2f(t * k);          // exp(S*t)
    const float num = SCALE * (t - MARGIN);
    const float excl = red[0] - et;                           // sum_{j!=y} exp(S*wf)
    const float denom = __builtin_amdgcn_exp2f(num * LOG2E) + excl;
    // log(denom) = ln2 * log2(denom)
    const float L = num - LN2 * __builtin_amdgcn_logf(denom);
    rowL[row] = L;
  }
}

// Single-wave reduction of NROWS fp32 values using chained f32 WMMA:
// D = A(16x4 data) x B(4x16 ones) + C  ==> every D[m][n] accumulates row sums.
__global__ __launch_bounds__(32) void adms_reduce_kernel(
    const float* __restrict__ rowL, float* __restrict__ out) {
  const int lane = threadIdx.x;                 // 0..31, EXEC all ones (no divergence)
  v8f c = {0.f, 0.f, 0.f, 0.f, 0.f, 0.f, 0.f, 0.f};
  const v2f ones = {1.0f, 1.0f};

  // 64 values per WMMA (A matrix), 4096 / 64 = 64 accumulating WMMAs.
  for (int i = 0; i < NROWS / 64; ++i) {
    v2f a = *(const v2f*)(rowL + i * 64 + lane * 2);
    c = __builtin_amdgcn_wmma_f32_16x16x4_f32(
        /*neg_a=*/false, a, /*neg_b=*/false, ones,
        /*c_mod=*/(short)0, c, /*reuse_a=*/false, /*reuse_b=*/false);
  }

  // Lane l (l<16) holds sum over M=0..7 at N=l; lane l+16 holds M=8..15 at N=l.
  float s = c[0] + c[1] + c[2] + c[3] + c[4] + c[5] + c[6] + c[7];
  s += __shfl_xor(s, 16, 32);                   // fold the two M-halves
  if (lane == 0) out[0] = -s * (1.0f / (float)NROWS);
}

extern "C" void kernel_launch(void* const* d_in, const int* in_sizes, int n_in,
                              void* d_out, int out_size, void* d_ws, size_t ws_size,
                              hipStream_t stream) {
  const float* wf = (const float*)d_in[0];
  const int* labels = (const int*)d_in[1];
  float* out = (float*)d_out;
  float* rowL = (float*)d_ws;                   // NROWS floats of scratch

  adms_row_kernel<<<NROWS, 256, 0, stream>>>(wf, labels, rowL);
  adms_reduce_kernel<<<1, 32, 0, stream>>>(rowL, out);
}